// BertEncoder_52613349376870
// MI455X (gfx1250) — compile-verified
//
#include <hip/hip_runtime.h>
#include <stdint.h>

#define BB 64
#define SS 512
#define DD 768
#define CHUNK 128              // floats of D per block
#define NCHUNK (DD / CHUNK)    // 6
#define RSTAGE 64              // rows per TDM stage (32KB per stage)
#define NSTAGE (SS / RSTAGE)   // 8
#define NTHREADS 128

typedef unsigned int v4u __attribute__((ext_vector_type(4)));
typedef int          v8i __attribute__((ext_vector_type(8)));
typedef int          v4i __attribute__((ext_vector_type(4)));

// ---- TDM: async DMA of a (RSTAGE x CHUNK) f32 tile (row stride DD floats)
// from global memory into LDS. Descriptor per CDNA5 ISA.
__device__ __forceinline__ void tdm_load_stage(const float* gsrc, unsigned lds_byte_off) {
  unsigned long long ga = (unsigned long long)(uintptr_t)gsrc;
  // Group 0: [1:0]=count=1 | lds_addr | global_addr[56:0] | type=2 (bits 127:126)
  v4u g0 = { 0x1u,
             lds_byte_off,
             (unsigned)(ga & 0xFFFFFFFFu),
             (unsigned)((ga >> 32) & 0x01FFFFFFu) | 0x80000000u };
  // Group 1 (constants, 8-byte data_size units):
  //   data_size=3 (8B); tensor_dim0=64; tensor_dim1=0xFFFF (OOB slack);
  //   tile_dim0=64 (=512B row), tile_dim1=RSTAGE, tile_dim2=0;
  //   tensor_dim0_stride=384 (=3072B = 768 floats); dim1 stride unused.
  v8i g1 = { (int)0x00030000,          // workgroup_mask=0, data_size=3
             (int)0x00400000,          // tensor_dim0[15:0]=64 in bits 63:48
             (int)0xFFFF0000,          // tensor_dim1[15:0]=0xFFFF
             (int)0x00400000,          // tile_dim0=64 in bits 127:112
             (int)RSTAGE,              // tile_dim1=RSTAGE, tile_dim2=0
             (int)((DD * 4) / 8),      // tensor_dim0_stride = 384
             0, 0 };
  v4i z4 = { 0, 0, 0, 0 };
  v8i z8 = { 0, 0, 0, 0, 0, 0, 0, 0 };
  // 6-arg form (clang-23 / therock-10.0 headers): groups 2/3 unused (2D tile).
  __builtin_amdgcn_tensor_load_to_lds(g0, g1, z4, z4, z8, 0);
}

// Scatter one (b, seg, d) value into the split (report, word) output layout.
__device__ __forceinline__ void store_out(float* out, int b, int segv, int d, float v) {
  size_t idx;
  if (segv == 0)
    idx = (size_t)b * DD + d;                                       // report_feat
  else
    idx = (size_t)BB * DD + ((size_t)(b * (SS - 1) + (segv - 1)) * DD + d); // word_feat
  __builtin_nontemporal_store(v, out + idx);
}

__global__ void __launch_bounds__(NTHREADS)
segment_sum_tdm_kernel(const float* __restrict__ emb,
                       const int* __restrict__ seg,
                       float* __restrict__ out) {
  __shared__ __align__(16) float buf[2][RSTAGE * CHUNK];   // 2 x 32KB double buffer
  __shared__ __align__(16) int   segl[SS];                 // 2KB segment ids

  const int tid   = threadIdx.x;
  const int b     = blockIdx.x / NCHUNK;
  const int chunk = blockIdx.x % NCHUNK;
  const int d     = chunk * CHUNK + tid;

  // Stage segment ids: exactly one int4 per thread (512 = 128*4), fully coalesced.
  {
    const int4* src = (const int4*)(seg + b * SS);
    ((int4*)segl)[tid] = src[tid];
  }

  const float* tile_base = emb + (size_t)b * SS * DD + chunk * CHUNK;

  // Uniform (scalar-branch) wave id: TDM issue must happen exactly once per block
  // (tensor ops ignore EXEC, so the guard must be a scalar branch).
  const unsigned wid  = ((unsigned)__builtin_amdgcn_readfirstlane((int)threadIdx.x)) >> 5;
  const unsigned lds0 = (unsigned)(uintptr_t)&buf[0][0];
  const unsigned lds1 = (unsigned)(uintptr_t)&buf[1][0];

  if (wid == 0) {
    tdm_load_stage(tile_base, lds0);                       // prefetch stage 0
  }

  float acc         = 0.0f;
  int   nextToWrite = 0;
  int   cur         = -1;                                  // loaded after first barrier

  for (int st = 0; st < NSTAGE; ++st) {
    if (wid == 0) {
      if (st + 1 < NSTAGE) {
        tdm_load_stage(tile_base + (size_t)(st + 1) * RSTAGE * DD,
                       ((st + 1) & 1) ? lds1 : lds0);      // prefetch next stage
        __builtin_amdgcn_s_wait_tensorcnt(1);              // current stage landed
      } else {
        __builtin_amdgcn_s_wait_tensorcnt(0);              // last stage landed
      }
    }
    __syncthreads();                                       // stage (and segl) visible
    if (st == 0) cur = segl[0];

    const float* cbuf = buf[st & 1];
#pragma unroll 4
    for (int r = 0; r < RSTAGE; ++r) {
      const int s = st * RSTAGE + r;
      acc += cbuf[r * CHUNK + tid];
      const int nxt = (s == SS - 1) ? -1 : segl[s + 1];    // one LDS read per row
      if (nxt != cur) {                                    // end of a sorted run
        for (int z = nextToWrite; z < cur; ++z)            // zero empty segments
          store_out(out, b, z, d, 0.0f);
        store_out(out, b, cur, d, acc);
        nextToWrite = cur + 1;
        acc = 0.0f;
      }
      cur = nxt;
    }
    __syncthreads();                                       // done reading this buffer
  }

  for (int z = nextToWrite; z < SS; ++z)                   // trailing empty segments
    store_out(out, b, z, d, 0.0f);
}

extern "C" void kernel_launch(void* const* d_in, const int* in_sizes, int n_in,
                              void* d_out, int out_size, void* d_ws, size_t ws_size,
                              hipStream_t stream) {
  (void)in_sizes; (void)n_in; (void)out_size; (void)d_ws; (void)ws_size;
  const float* emb = (const float*)d_in[0];
  const int*   seg = (const int*)d_in[1];
  float*       out = (float*)d_out;

  dim3 grid(BB * NCHUNK);   // 384 blocks
  dim3 block(NTHREADS);     // 128 threads = 4 waves
  segment_sum_tdm_kernel<<<grid, block, 0, stream>>>(emb, seg, out);
}